// SMoE_46935402611077
// MI455X (gfx1250) — compile-verified
//
#include <hip/hip_runtime.h>
#include <cstddef>

// ---------------- problem constants ----------------
constexpr int N_TOK = 8192;
constexpr int D_IN  = 1024;
constexpr int H_DIM = 1024;
constexpr int D_OUT = 1024;
constexpr int NEXP  = 8;
constexpr float CAP = 1.0f - 0.2f;   // 1 - EPS

typedef float v2f __attribute__((ext_vector_type(2)));
typedef float v8f __attribute__((ext_vector_type(8)));

// ---------------------------------------------------------------------------
// Kernel 0: zero output + expert counters (harness poisons buffers with 0xAA)
// ---------------------------------------------------------------------------
__global__ __launch_bounds__(256) void smoe_zero(float* __restrict__ out,
                                                 int* __restrict__ counts,
                                                 size_t n4) {
  size_t i = (size_t)blockIdx.x * blockDim.x + threadIdx.x;
  size_t stride = (size_t)gridDim.x * blockDim.x;
  float4 z = make_float4(0.f, 0.f, 0.f, 0.f);
  float4* o4 = reinterpret_cast<float4*>(out);
  for (; i < n4; i += stride) o4[i] = z;
  if (blockIdx.x == 0 && threadIdx.x < NEXP) counts[threadIdx.x] = 0;
}

// ---------------------------------------------------------------------------
// Kernel 1: router. One thread per token.
//   logits = x @ Wsel + bsel ; softmax ; sort desc ; capped-cumsum sparse
//   weights ; replicate reference's gather quirk sw[e] = s_sorted[idx[e]] ;
//   append active tokens to per-expert lists.
// ---------------------------------------------------------------------------
__global__ __launch_bounds__(256) void smoe_route(const float* __restrict__ x,
                                                  const float* __restrict__ Wsel,
                                                  const float* __restrict__ bsel,
                                                  float* __restrict__ sw,
                                                  int* __restrict__ counts,
                                                  int* __restrict__ tokidx) {
  __shared__ float wsel_s[D_IN * NEXP];   // 32 KB
  int tid = threadIdx.x;
  for (int i = tid; i < D_IN * NEXP; i += 256) wsel_s[i] = Wsel[i];
  __syncthreads();

  int n = blockIdx.x * 256 + tid;
  const float* xr = x + (size_t)n * D_IN;

  float acc[NEXP];
#pragma unroll
  for (int e = 0; e < NEXP; ++e) acc[e] = bsel[e];

  for (int d = 0; d < D_IN; ++d) {
    float xv = xr[d];
#pragma unroll
    for (int e = 0; e < NEXP; ++e) acc[e] = fmaf(xv, wsel_s[d * NEXP + e], acc[e]);
  }

  // softmax
  float mx = acc[0];
#pragma unroll
  for (int e = 1; e < NEXP; ++e) mx = fmaxf(mx, acc[e]);
  float sum = 0.f;
#pragma unroll
  for (int e = 0; e < NEXP; ++e) { acc[e] = __expf(acc[e] - mx); sum += acc[e]; }
  float inv = 1.f / sum;
#pragma unroll
  for (int e = 0; e < NEXP; ++e) acc[e] *= inv;

  // sort descending, tracking original expert index
  int idx[NEXP];
#pragma unroll
  for (int e = 0; e < NEXP; ++e) idx[e] = e;
#pragma unroll
  for (int i = 0; i < NEXP - 1; ++i) {
#pragma unroll
    for (int j = 0; j < NEXP - 1 - i; ++j) {
      if (acc[j] < acc[j + 1]) {
        float tw = acc[j]; acc[j] = acc[j + 1]; acc[j + 1] = tw;
        int   ti = idx[j]; idx[j] = idx[j + 1]; idx[j + 1] = ti;
      }
    }
  }

  // capped cumulative sum -> sparse weights in sorted order
  float ssort[NEXP];
  float cum = 0.f;
#pragma unroll
  for (int j = 0; j < NEXP; ++j) {
    float w = acc[j];
    cum += w;
    float capped = fminf(cum, CAP);
    ssort[j] = fmaxf(capped - cum + w, 0.f);
  }

  // faithful to reference: sw[n][e] = ssort[ idx[e] ]  (gather, not scatter)
#pragma unroll
  for (int e = 0; e < NEXP; ++e) {
    float val = ssort[idx[e]];
    sw[(size_t)n * NEXP + e] = val;
    if (val > 0.f) {
      int pos = atomicAdd(&counts[e], 1);
      tokidx[(size_t)e * N_TOK + pos] = n;
    }
  }
}

// ---------------------------------------------------------------------------
// Kernel 2: fused 2-layer expert MLP over routed token tiles.
//   Block = (expert e, tile of 32 tokens), 256 threads = 8 waves.
//   Phase A: gather 32 x-rows into LDS.
//   Phase B: h = relu(x @ W1_e + b1_e)  -> LDS (32 x 1024)
//   Phase C: y = h @ W2_e + b2_e ; out[tok] += sw * y  (fp32 atomics)
//   Each wave owns 128 N-columns and carries TWO 16x16 accumulator tiles
//   (token rows 0-15 and 16-31) so every B fragment feeds two WMMAs:
//   B-operand L2 traffic per FLOP is halved vs a 16-token tile.
// ---------------------------------------------------------------------------
constexpr int TM     = 32;    // tokens per block tile
constexpr int XPITCH = 1028;  // row pitch pad: 1028*4 B -> 16 rows hit 16 banks

__global__ __launch_bounds__(256) void smoe_experts(const float* __restrict__ x,
                                                    const float* __restrict__ W1,
                                                    const float* __restrict__ b1,
                                                    const float* __restrict__ W2,
                                                    const float* __restrict__ b2,
                                                    const float* __restrict__ sw,
                                                    const int* __restrict__ counts,
                                                    const int* __restrict__ tokidx,
                                                    float* __restrict__ out) {
  __shared__ __align__(16) float xs[TM * XPITCH];  // 128.5 KB gathered x rows
  __shared__ __align__(16) float hs[TM * XPITCH];  // 128.5 KB hidden tile
  __shared__ int   tok[TM];
  __shared__ float swv[TM];

  const int bx = blockIdx.x;
  const int e  = bx >> 8;        // 256 tiles of 32 tokens per expert
  const int t  = bx & 255;
  const int cnt = counts[e];
  if (t * TM >= cnt) return;     // block-uniform: EXEC stays all-ones for WMMA

  const int tid  = threadIdx.x;
  const int lane = tid & 31;
  const int wv   = tid >> 5;               // wave id 0..7
  const int mA   = lane & 15;              // A-operand row (M) within 16-tile
  const int koff = (lane >> 4) << 1;       // K sub-offset: 0 (lanes 0-15) / 2 (16-31)
  const int mHi  = (lane < 16) ? 0 : 8;    // C/D row offset per lane half

  if (tid < TM) {
    int slot = t * TM + tid;
    int tk   = (slot < cnt) ? tokidx[(size_t)e * N_TOK + slot] : 0;
    tok[tid] = tk;
    swv[tid] = (slot < cnt) ? sw[(size_t)tk * NEXP + e] : 0.f;
  }
  __syncthreads();

  // ---- Phase A: stage 32 gathered rows (32 x 1024 f32), float4 loads ----
#pragma unroll
  for (int it = 0; it < 32; ++it) {
    int idx4 = tid + it * 256;       // 0..8191 float4 slots
    int m    = idx4 >> 8;            // row
    int c4   = (idx4 & 255) << 2;    // column
    float4 v = *reinterpret_cast<const float4*>(x + (size_t)tok[m] * D_IN + c4);
    float* dst = &xs[m * XPITCH + c4];
    dst[0] = v.x; dst[1] = v.y; dst[2] = v.z; dst[3] = v.w;
  }
  __syncthreads();

  // ---- Phase B: h = relu(x @ W1_e + b1_e) ----
  const float* W1e = W1 + (size_t)e * D_IN * H_DIM;
  const float* b1e = b1 + (size_t)e * H_DIM;
#pragma unroll 1
  for (int nt = 0; nt < 8; ++nt) {
    const int n = wv * 128 + nt * 16 + (lane & 15);
    const float* bcol = W1e + n;
    v8f acc0 = {};   // token rows 0-15
    v8f acc1 = {};   // token rows 16-31
#pragma unroll 4
    for (int k0 = 0; k0 < D_IN; k0 += 4) {
      const int kk = k0 + koff;
      v2f a0, a1, b;
      a0.x = xs[mA * XPITCH + kk];
      a0.y = xs[mA * XPITCH + kk + 1];
      a1.x = xs[(16 + mA) * XPITCH + kk];
      a1.y = xs[(16 + mA) * XPITCH + kk + 1];
      b.x  = bcol[(size_t)kk * H_DIM];
      b.y  = bcol[(size_t)(kk + 1) * H_DIM];
      acc0 = __builtin_amdgcn_wmma_f32_16x16x4_f32(false, a0, false, b,
                                                   (short)0, acc0, false, false);
      acc1 = __builtin_amdgcn_wmma_f32_16x16x4_f32(false, a1, false, b,
                                                   (short)0, acc1, false, false);
    }
    const float bias = b1e[n];
#pragma unroll
    for (int i = 0; i < 8; ++i) {
      float h0 = acc0[i] + bias;
      float h1 = acc1[i] + bias;
      hs[(i + mHi) * XPITCH + n]      = fmaxf(h0, 0.f);
      hs[(16 + i + mHi) * XPITCH + n] = fmaxf(h1, 0.f);
    }
  }
  __syncthreads();

  // ---- Phase C: y = h @ W2_e + b2_e ; scatter sw*y into out ----
  const float* W2e = W2 + (size_t)e * H_DIM * D_OUT;
  const float* b2e = b2 + (size_t)e * D_OUT;
#pragma unroll 1
  for (int nt = 0; nt < 8; ++nt) {
    const int n = wv * 128 + nt * 16 + (lane & 15);
    const float* bcol = W2e + n;
    v8f acc0 = {};
    v8f acc1 = {};
#pragma unroll 4
    for (int k0 = 0; k0 < H_DIM; k0 += 4) {
      const int kk = k0 + koff;
      v2f a0, a1, b;
      a0.x = hs[mA * XPITCH + kk];
      a0.y = hs[mA * XPITCH + kk + 1];
      a1.x = hs[(16 + mA) * XPITCH + kk];
      a1.y = hs[(16 + mA) * XPITCH + kk + 1];
      b.x  = bcol[(size_t)kk * D_OUT];
      b.y  = bcol[(size_t)(kk + 1) * D_OUT];
      acc0 = __builtin_amdgcn_wmma_f32_16x16x4_f32(false, a0, false, b,
                                                   (short)0, acc0, false, false);
      acc1 = __builtin_amdgcn_wmma_f32_16x16x4_f32(false, a1, false, b,
                                                   (short)0, acc1, false, false);
    }
    const float bias = b2e[n];
#pragma unroll
    for (int i = 0; i < 8; ++i) {
      int m0 = i + mHi;
      int m1 = 16 + i + mHi;
      float y0 = acc0[i] + bias;
      float y1 = acc1[i] + bias;
      atomicAdd(out + (size_t)tok[m0] * D_OUT + n, swv[m0] * y0);
      atomicAdd(out + (size_t)tok[m1] * D_OUT + n, swv[m1] * y1);
    }
  }
}

// ---------------------------------------------------------------------------
// Host-side launcher
// ---------------------------------------------------------------------------
extern "C" void kernel_launch(void* const* d_in, const int* in_sizes, int n_in,
                              void* d_out, int out_size, void* d_ws, size_t ws_size,
                              hipStream_t stream) {
  const float* x    = (const float*)d_in[0];
  const float* Wsel = (const float*)d_in[1];
  const float* bsel = (const float*)d_in[2];
  const float* W1   = (const float*)d_in[3];
  const float* b1   = (const float*)d_in[4];
  const float* W2   = (const float*)d_in[5];
  const float* b2   = (const float*)d_in[6];
  float* out = (float*)d_out;

  // workspace layout: [counts: 8 ints, padded to 256B][sw: N*E f32][tokidx: E*N i32]
  char* ws = (char*)d_ws;
  int*   counts = (int*)ws;
  float* sw     = (float*)(ws + 256);
  int*   tokidx = (int*)(ws + 256 + (size_t)N_TOK * NEXP * sizeof(float));

  size_t out_f4 = ((size_t)N_TOK * D_OUT) / 4;
  smoe_zero<<<2048, 256, 0, stream>>>(out, counts, out_f4);
  smoe_route<<<N_TOK / 256, 256, 0, stream>>>(x, Wsel, bsel, sw, counts, tokidx);
  smoe_experts<<<NEXP * (N_TOK / TM), 256, 0, stream>>>(x, W1, b1, W2, b2,
                                                        sw, counts, tokidx, out);
}